// A2C_12884901888487
// MI455X (gfx1250) — compile-verified
//
#include <hip/hip_runtime.h>
#include <math.h>

typedef __attribute__((ext_vector_type(2))) float v2f;
typedef __attribute__((ext_vector_type(8))) float v8f;

#define CDIM 128
#define HDIM 32

__device__ __forceinline__ float softplusf(float x) {
    return fmaxf(x, 0.0f) + log1pf(expf(-fabsf(x)));
}

// ---------------- init: deg=1 (self loop), hsum=0, psum=0 ----------------
__global__ void init_kernel(float* __restrict__ deg, float* __restrict__ hsum,
                            float* __restrict__ psum, int n) {
    int i = blockIdx.x * blockDim.x + threadIdx.x;
    if (i < n) deg[i] = 1.0f;
    if (i < CDIM) hsum[i] = 0.0f;
    if (i == 0) psum[0] = 0.0f;
}

// ---------------- degree accumulation over edges ----------------
__global__ void deg_kernel(const int* __restrict__ dst, float* __restrict__ deg, int E) {
    int e = blockIdx.x * blockDim.x + threadIdx.x;
    if (e < E) atomicAdd(&deg[dst[e]], 1.0f);
}

__global__ void dinv_kernel(const float* __restrict__ deg, float* __restrict__ dinv, int n) {
    int i = blockIdx.x * blockDim.x + threadIdx.x;
    if (i < n) {
        float d = deg[i];
        dinv[i] = (d > 0.0f) ? rsqrtf(d) : 0.0f;
    }
}

// ---- fp32 WMMA GEMM, one wave per 16x32 output slab (2 accumulators share A) ----
// Out = [relu](A[rows,K] @ B[K,ncols] + bias)
// Optional fused self-loop epilogue: AggOut[r][c] = val * dinv[r]^2
__global__ __launch_bounds__(256) void wmma_gemm2_kernel(
    const float* __restrict__ A, const float* __restrict__ B,
    const float* __restrict__ bias, float* __restrict__ Out,
    const float* __restrict__ dinv, float* __restrict__ AggOut,
    int rows, int K, int ncols, int doRelu)
{
    int warp = threadIdx.x >> 5;
    int lane = threadIdx.x & 31;
    int colPairs = ncols >> 5;                       // 32-wide column slabs
    int tile = blockIdx.x * (blockDim.x >> 5) + warp;
    int totalTiles = (rows >> 4) * colPairs;
    if (tile >= totalTiles) return;                  // wave-uniform: EXEC stays full
    int rowTile = tile / colPairs;
    int cp = tile - rowTile * colPairs;
    int row0 = rowTile << 4;
    int col0 = cp << 5;
    int half = lane >> 4;                            // A: K+2h,K+2h+1 ; lanes hold M=0..15
    int lm   = lane & 15;

    v8f acc0 = {0.f,0.f,0.f,0.f,0.f,0.f,0.f,0.f};
    v8f acc1 = {0.f,0.f,0.f,0.f,0.f,0.f,0.f,0.f};
    const float* arow = A + (long long)(row0 + lm) * K + 2 * half;
    const float* b0p  = B + (long long)(2 * half) * ncols + col0 + lm;
    const float* b1p  = b0p + 16;

    for (int k = 0; k < K; k += 4) {
        v2f a, b0, b1;
        a.x  = arow[0];
        a.y  = arow[1];
        b0.x = b0p[0];
        b0.y = b0p[ncols];
        b1.x = b1p[0];
        b1.y = b1p[ncols];
        acc0 = __builtin_amdgcn_wmma_f32_16x16x4_f32(
            false, a, false, b0, (short)0, acc0, false, false);
        acc1 = __builtin_amdgcn_wmma_f32_16x16x4_f32(
            false, a, false, b1, (short)0, acc1, false, false);
        arow += 4;
        b0p  += 4 * ncols;
        b1p  += 4 * ncols;
    }

    float bv0 = 0.0f, bv1 = 0.0f;
    if (bias != nullptr) {
        bv0 = bias[col0 + lm];
        bv1 = bias[col0 + 16 + lm];
    }
    long long obase = (long long)(row0 + 8 * half) * ncols + col0 + lm;
    float* orow = Out + obase;
#pragma unroll
    for (int r = 0; r < 8; ++r) {
        float v0 = acc0[r] + bv0;
        float v1 = acc1[r] + bv1;
        if (doRelu) {
            v0 = fmaxf(v0, 0.0f);
            v1 = fmaxf(v1, 0.0f);
        }
        long long off = (long long)r * ncols;
        orow[off]      = v0;
        orow[off + 16] = v1;
        if (AggOut != nullptr) {                     // fused self-loop: agg = val*dinv^2
            float di = dinv[row0 + 8 * half + r];
            float w = di * di;
            float* g = AggOut + obase;
            g[off]      = v0 * w;
            g[off + 16] = v1 * w;
        }
    }
}

// ---------------- edge scatter: one wave per edge, both convs in one pass -----------
__global__ __launch_bounds__(256) void scatter_kernel(
    const int* __restrict__ src, const int* __restrict__ dst,
    const float* __restrict__ dinv,
    const float* __restrict__ ha, const float* __restrict__ hc,
    float* __restrict__ agga, float* __restrict__ aggc, int E)
{
    int e = blockIdx.x * (blockDim.x >> 5) + (threadIdx.x >> 5);
    int lane = threadIdx.x & 31;
    if (e >= E) return;
    int s = src[e], d = dst[e];
    float nrm = dinv[s] * dinv[d];
    const float4 va = ((const float4*)(ha + (long long)s * CDIM))[lane];
    const float4 vc = ((const float4*)(hc + (long long)s * CDIM))[lane];
    float* oa = agga + (long long)d * CDIM + lane * 4;
    float* oc = aggc + (long long)d * CDIM + lane * 4;
    atomicAdd(oa + 0, va.x * nrm);
    atomicAdd(oa + 1, va.y * nrm);
    atomicAdd(oa + 2, va.z * nrm);
    atomicAdd(oa + 3, va.w * nrm);
    atomicAdd(oc + 0, vc.x * nrm);
    atomicAdd(oc + 1, vc.y * nrm);
    atomicAdd(oc + 2, vc.z * nrm);
    atomicAdd(oc + 3, vc.w * nrm);
}

// ------ fused: h = relu(agg + bias) + x  (in place), plus critic column sums --------
__global__ __launch_bounds__(128) void fuse_kernel(
    const float* __restrict__ x, const float* __restrict__ ab, const float* __restrict__ cb,
    float* __restrict__ agga, float* __restrict__ aggc, float* __restrict__ hsum,
    int N, int rowsPerBlock)
{
    int c = threadIdx.x;  // 128 threads = one per column
    long long r0 = (long long)blockIdx.x * rowsPerBlock;
    long long r1 = r0 + rowsPerBlock;
    if (r1 > N) r1 = N;
    float ba = ab[c], bc = cb[c], sum = 0.0f;
    for (long long r = r0; r < r1; ++r) {
        long long idx = r * CDIM + c;
        float xv = x[idx];
        float va = fmaxf(agga[idx] + ba, 0.0f) + xv;
        agga[idx] = va;
        float vc = fmaxf(aggc[idx] + bc, 0.0f) + xv;
        aggc[idx] = vc;
        sum += vc;
    }
    atomicAdd(&hsum[c], sum);
}

// ---------------- actor head: conc = softplus(a0)+J, p = softplus(a1), sum p --------
__global__ __launch_bounds__(256) void head_kernel(
    const float* __restrict__ t2, const float* __restrict__ a3W, const float* __restrict__ a3b,
    float* __restrict__ out_conc, float* __restrict__ pbuf, float* __restrict__ psum, int N)
{
    __shared__ float red[256];
    int i = blockIdx.x * blockDim.x + threadIdx.x;
    float p = 0.0f;
    if (i < N) {
        const float* row = t2 + (long long)i * HDIM;
        float a0 = a3b[0], a1 = a3b[1];
#pragma unroll
        for (int k = 0; k < HDIM; ++k) {
            float v = row[k];
            a0 += v * a3W[k * 2 + 0];
            a1 += v * a3W[k * 2 + 1];
        }
        out_conc[i] = softplusf(a0) + 1e-20f;
        p = softplusf(a1);
        pbuf[i] = p;
    }
    red[threadIdx.x] = p;
    __syncthreads();
    for (int s = blockDim.x >> 1; s > 0; s >>= 1) {
        if ((int)threadIdx.x < s) red[threadIdx.x] += red[threadIdx.x + s];
        __syncthreads();
    }
    if (threadIdx.x == 0) atomicAdd(psum, red[0]);
}

// ---------------- critic MLP on summed vector (single block, 32 threads) ------------
__global__ void critic_kernel(const float* __restrict__ hsum,
                              const float* __restrict__ c1W, const float* __restrict__ c1b,
                              const float* __restrict__ c2W, const float* __restrict__ c2b,
                              const float* __restrict__ c3W, const float* __restrict__ c3b,
                              float* __restrict__ out_value)
{
    __shared__ float t1[HDIM], t2s[HDIM];
    int j = threadIdx.x;
    float acc = c1b[j];
    for (int k = 0; k < CDIM; ++k) acc += hsum[k] * c1W[k * HDIM + j];
    t1[j] = fmaxf(acc, 0.0f);
    __syncthreads();
    acc = c2b[j];
    for (int k = 0; k < HDIM; ++k) acc += t1[k] * c2W[k * HDIM + j];
    t2s[j] = fmaxf(acc, 0.0f);
    __syncthreads();
    if (j < 2) {
        acc = c3b[j];
        for (int k = 0; k < HDIM; ++k) acc += t2s[k] * c3W[k * 2 + j];
        out_value[j] = acc;
    }
}

// ---------------- normalize action2_probs ----------------
__global__ void norm_kernel(const float* __restrict__ pbuf, const float* __restrict__ psum,
                            float* __restrict__ out_probs, int N)
{
    int i = blockIdx.x * blockDim.x + threadIdx.x;
    if (i < N) out_probs[i] = pbuf[i] / psum[0];
}

extern "C" void kernel_launch(void* const* d_in, const int* in_sizes, int n_in,
                              void* d_out, int out_size, void* d_ws, size_t ws_size,
                              hipStream_t stream)
{
    // inputs (setup_inputs order)
    const float* x    = (const float*)d_in[0];
    const int*   ei   = (const int*)d_in[1];
    const float* aW   = (const float*)d_in[2];
    const float* ab   = (const float*)d_in[3];
    const float* a1W  = (const float*)d_in[4];
    const float* a1b  = (const float*)d_in[5];
    const float* a2W  = (const float*)d_in[6];
    const float* a2b  = (const float*)d_in[7];
    const float* a3W  = (const float*)d_in[8];
    const float* a3b  = (const float*)d_in[9];
    const float* cW   = (const float*)d_in[10];
    const float* cb   = (const float*)d_in[11];
    const float* c1W  = (const float*)d_in[12];
    const float* c1b  = (const float*)d_in[13];
    const float* c2W  = (const float*)d_in[14];
    const float* c2b  = (const float*)d_in[15];
    const float* c3W  = (const float*)d_in[16];
    const float* c3b  = (const float*)d_in[17];

    const int N = in_sizes[0] / CDIM;     // 50000
    const int E = in_sizes[1] / 2;        // 1.6M
    const int* src = ei;
    const int* dst = ei + E;

    float* out = (float*)d_out;
    float* out_conc  = out;               // [0, N)
    float* out_value = out + N;           // [N, N+2)
    float* out_probs = out + N + 2;       // [N+2, 2N+2)

    // workspace layout
    float* ws     = (float*)d_ws;
    float* deg    = ws;                         // N
    float* dinv   = deg + N;                    // N
    float* pbuf   = dinv + N;                   // N
    float* hsum   = pbuf + N;                   // 128
    float* psum   = hsum + CDIM;                // 1 (+pad)
    float* ha_pre = psum + 8;                   // N*128
    float* hc_pre = ha_pre + (long long)N * CDIM;
    float* agga   = hc_pre + (long long)N * CDIM;
    float* aggc   = agga + (long long)N * CDIM;
    float* t1     = aggc + (long long)N * CDIM; // N*32
    float* t2     = t1 + (long long)N * HDIM;   // N*32

    // 1. init deg=1 (self loop), hsum=0, psum=0
    init_kernel<<<(N + 255) / 256, 256, 0, stream>>>(deg, hsum, psum, N);
    // 2. degree over edges
    deg_kernel<<<(E + 255) / 256, 256, 0, stream>>>(dst, deg, E);
    // 3. dinv = rsqrt(deg)
    dinv_kernel<<<(N + 255) / 256, 256, 0, stream>>>(deg, dinv, N);
    // 4. conv GEMMs with fused self-loop epilogue:
    //    ha_pre = x @ aW, agga = ha_pre*dinv^2 ; same for critic
    {
        int tiles = (N >> 4) * (CDIM >> 5);          // 16x32 slabs
        int blocks = (tiles + 7) / 8;
        wmma_gemm2_kernel<<<blocks, 256, 0, stream>>>(x, aW, nullptr, ha_pre,
                                                      dinv, agga, N, CDIM, CDIM, 0);
        wmma_gemm2_kernel<<<blocks, 256, 0, stream>>>(x, cW, nullptr, hc_pre,
                                                      dinv, aggc, N, CDIM, CDIM, 0);
    }
    // 5. edge scatter with symmetric norm (both convs in one pass)
    scatter_kernel<<<(E + 7) / 8, 256, 0, stream>>>(src, dst, dinv, ha_pre, hc_pre, agga, aggc, E);
    // 6. fused relu+bias+residual, critic column sums
    {
        const int rowsPerBlock = 64;
        int blocks = (N + rowsPerBlock - 1) / rowsPerBlock;
        fuse_kernel<<<blocks, 128, 0, stream>>>(x, ab, cb, agga, aggc, hsum, N, rowsPerBlock);
    }
    // 7. actor MLP layer 1: t1 = relu(h @ a1W + a1b)   [N,128]@[128,32]
    {
        int tiles = (N >> 4) * (HDIM >> 5);          // one 16x32 slab per 16 rows
        int blocks = (tiles + 7) / 8;
        wmma_gemm2_kernel<<<blocks, 256, 0, stream>>>(agga, a1W, a1b, t1,
                                                      nullptr, nullptr, N, CDIM, HDIM, 1);
        // 8. actor MLP layer 2: t2 = relu(t1 @ a2W + a2b)   [N,32]@[32,32]
        wmma_gemm2_kernel<<<blocks, 256, 0, stream>>>(t1, a2W, a2b, t2,
                                                      nullptr, nullptr, N, HDIM, HDIM, 1);
    }
    // 9. actor head: concentration + p + sum(p)
    head_kernel<<<(N + 255) / 256, 256, 0, stream>>>(t2, a3W, a3b, out_conc, pbuf, psum, N);
    // 10. critic MLP on column sums
    critic_kernel<<<1, HDIM, 0, stream>>>(hsum, c1W, c1b, c2W, c2b, c3W, c3b, out_value);
    // 11. normalize probs
    norm_kernel<<<(N + 255) / 256, 256, 0, stream>>>(pbuf, psum, out_probs, N);
}